// DenseFusionLoss_28741921144894
// MI455X (gfx1250) — compile-verified
//
#include <hip/hip_runtime.h>
#include <math.h>

typedef __attribute__((ext_vector_type(2))) float v2f;
typedef __attribute__((ext_vector_type(8))) float v8f;

#define VNUM 2048          // vertices per model (reference V)
#define KTILES (VNUM / 16) // 128 gt tiles of 16 points

__device__ __forceinline__ void quat_pose_to_rt(const float* __restrict__ pose,
                                                float R[9], float t[3]) {
    t[0] = pose[0]; t[1] = pose[1]; t[2] = pose[2];
    float w = pose[3], x = pose[4], y = pose[5], z = pose[6];
    float inv = rsqrtf(w * w + x * x + y * y + z * z);
    w *= inv; x *= inv; y *= inv; z *= inv;
    float xx = x * x, yy = y * y, zz = z * z;
    float xy = x * y, xz = x * z, yz = y * z;
    float wx = w * x, wy = w * y, wz = w * z;
    R[0] = 1.0f - 2.0f * (yy + zz); R[1] = 2.0f * (xy - wz);        R[2] = 2.0f * (xz + wy);
    R[3] = 2.0f * (xy + wz);        R[4] = 1.0f - 2.0f * (xx + zz); R[5] = 2.0f * (yz - wx);
    R[6] = 2.0f * (xz - wy);        R[7] = 2.0f * (yz + wx);        R[8] = 1.0f - 2.0f * (xx + yy);
}

__global__ void init_ws_kernel(float* ws, int n) {
    int i = blockIdx.x * blockDim.x + threadIdx.x;
    if (i < n) ws[i] = 0.0f;
}

// One block = 1/16 of one batch's pred rows (128 rows). 8 waves, each owning a
// 16-row pred tile. Full gt point set for the batch is staged in LDS,
// pre-formatted as WMMA B-fragments:
//   sB[0][v] = (gx, gy)        read by lanes 0-15  (K = 0,1)
//   sB[1][v] = (gz, |g|^2)     read by lanes 16-31 (K = 2,3)
__global__ void __launch_bounds__(256)
addloss_kernel(const float* __restrict__ pred_poses,
               const float* __restrict__ gt_poses,
               const float* __restrict__ model_vertices,
               const int* __restrict__ class_ids,
               float* __restrict__ add_sum,     // [B] sum of corresponding dists
               float* __restrict__ adds_sum) {  // [B] sum of min dists (ADD-S)
    __shared__ v2f   sB[2][VNUM];       // B fragments          32 KB
    __shared__ float sMin[8][16][16];   // per-wave min tiles    8 KB

    const int batch = blockIdx.x >> 4;
    const int chunk = blockIdx.x & 15;
    const int tid   = threadIdx.x;
    const int lane  = tid & 31;
    const int wave  = tid >> 5;
    const int m     = lane & 15;     // local row (A) / local col (B)
    const int hi    = lane >> 4;     // 0: K={0,1}, 1: K={2,3}

    float Rp[9], tp[3], Rg[9], tg[3];
    quat_pose_to_rt(pred_poses + batch * 7, Rp, tp);
    quat_pose_to_rt(gt_poses + batch * 7, Rg, tg);

    const float* verts = model_vertices + (size_t)class_ids[batch] * VNUM * 3;

    // Stage all gt-transformed points for this batch into LDS (fragment layout).
    for (int v = tid; v < VNUM; v += 256) {
        float x = verts[3 * v + 0], y = verts[3 * v + 1], z = verts[3 * v + 2];
        float gx = Rg[0] * x + Rg[1] * y + Rg[2] * z + tg[0];
        float gy = Rg[3] * x + Rg[4] * y + Rg[5] * z + tg[1];
        float gz = Rg[6] * x + Rg[7] * y + Rg[8] * z + tg[2];
        v2f lo; lo[0] = gx; lo[1] = gy;
        v2f hi2; hi2[0] = gz; hi2[1] = gx * gx + gy * gy + gz * gz;
        sB[0][v] = lo;
        sB[1][v] = hi2;
    }
    __syncthreads();

    // This wave's pred row tile: rows rt*16 .. rt*16+15.
    const int rt   = chunk * 8 + wave;
    const int vrow = rt * 16 + m;
    float x = verts[3 * vrow + 0], y = verts[3 * vrow + 1], z = verts[3 * vrow + 2];
    float px = Rp[0] * x + Rp[1] * y + Rp[2] * z + tp[0];
    float py = Rp[3] * x + Rp[4] * y + Rp[5] * z + tp[1];
    float pz = Rp[6] * x + Rp[7] * y + Rp[8] * z + tp[2];
    float pn = px * px + py * py + pz * pz;

    // A fragment (16x4 f32): lanes 0-15 hold K=0,1; lanes 16-31 hold K=2,3.
    // Row M: (-2px, -2py, -2pz, 1)
    v2f afrag;
    afrag[0] = (hi == 0) ? (-2.0f * px) : (-2.0f * pz);
    afrag[1] = (hi == 0) ? (-2.0f * py) : 1.0f;

    v8f runmin;
#pragma unroll
    for (int j = 0; j < 8; ++j) runmin[j] = 3.0e38f;

    const v8f czero = {0.f, 0.f, 0.f, 0.f, 0.f, 0.f, 0.f, 0.f};
    const v2f* __restrict__ bptr = &sB[hi][m];   // per-lane B-fragment base

#pragma unroll 4
    for (int t = 0; t < KTILES; ++t) {
        // B fragment (4x16 f32), col N = gt point t*16+N: (gx,gy,gz,|g|^2)
        v2f bfrag = bptr[t * 16];
        // e[M][N] = -2*dot(p_M, g_N) + |g_N|^2
        v8f e = __builtin_amdgcn_wmma_f32_16x16x4_f32(
            false, afrag, false, bfrag, (short)0, czero, false, false);
#pragma unroll
        for (int j = 0; j < 8; ++j) runmin[j] = fminf(runmin[j], e[j]);
    }

    // Stage mins: D layout -> row M = j + 8*hi, col N = lane&15.
#pragma unroll
    for (int j = 0; j < 8; ++j) sMin[wave][j + 8 * hi][m] = runmin[j];
    __syncthreads();

    float contrib_s = 0.0f, contrib_a = 0.0f;
    if (lane < 16) {
        float mn = 3.0e38f;
#pragma unroll
        for (int n = 0; n < 16; ++n) mn = fminf(mn, sMin[wave][lane][n]);
        float d2 = mn + pn;                        // min_w ||p-g||^2
        contrib_s = sqrtf(fmaxf(d2, 1e-12f));
        // Corresponding gt point, recomputed from registers.
        float gx = Rg[0] * x + Rg[1] * y + Rg[2] * z + tg[0];
        float gy = Rg[3] * x + Rg[4] * y + Rg[5] * z + tg[1];
        float gz = Rg[6] * x + Rg[7] * y + Rg[8] * z + tg[2];
        float dx = px - gx, dy = py - gy, dz = pz - gz;
        contrib_a = sqrtf(dx * dx + dy * dy + dz * dz);
    }
#pragma unroll
    for (int off = 16; off >= 1; off >>= 1) {
        contrib_s += __shfl_xor(contrib_s, off, 32);
        contrib_a += __shfl_xor(contrib_a, off, 32);
    }
    if (lane == 0) {
        atomicAdd(&adds_sum[batch], contrib_s);
        atomicAdd(&add_sum[batch], contrib_a);
    }
}

__global__ void conf_kernel(const float* __restrict__ conf, float* __restrict__ conf_sum, int n) {
    int i = blockIdx.x * blockDim.x + threadIdx.x;
    float v = 0.0f;
    if (i < n) {
        float zq = -conf[i];                              // softplus(-x)
        v = fmaxf(zq, 0.0f) + log1pf(expf(-fabsf(zq)));
    }
#pragma unroll
    for (int off = 16; off >= 1; off >>= 1) v += __shfl_xor(v, off, 32);
    if ((threadIdx.x & 31) == 0) atomicAdd(conf_sum, v);
}

__global__ void finalize_kernel(const float* __restrict__ pred_poses,
                                const int* __restrict__ class_ids,
                                const int* __restrict__ sym_mask,
                                const float* __restrict__ add_sum,
                                const float* __restrict__ adds_sum,
                                const float* __restrict__ conf_sum,
                                float* __restrict__ out, int B, int nconf) {
    int b = threadIdx.x;
    float sel = 0.0f, pr = 0.0f;
    if (b < B) {
        bool sym = sym_mask[class_ids[b]] > 0;
        float a  = add_sum[b]  * (1.0f / (float)VNUM);
        float as = adds_sum[b] * (1.0f / (float)VNUM);
        sel = sym ? as : a;
        float tx = pred_poses[b * 7 + 0];
        float ty = pred_poses[b * 7 + 1];
        float tz = pred_poses[b * 7 + 2];
        float tn = sqrtf(tx * tx + ty * ty + tz * tz);
        float e  = fmaxf(tn - 2.0f, 0.0f);
        pr = e * e;
    }
#pragma unroll
    for (int off = 16; off >= 1; off >>= 1) {
        sel += __shfl_xor(sel, off, 32);
        pr  += __shfl_xor(pr, off, 32);
    }
    if (threadIdx.x == 0) {
        float avg_add  = sel / (float)B;
        float conf_ls  = conf_sum[0] / (float)nconf;
        float pose_reg = pr / (float)B;
        out[0] = 1.0f * avg_add + 0.1f * conf_ls + 0.1f * pose_reg;
    }
}

extern "C" void kernel_launch(void* const* d_in, const int* in_sizes, int n_in,
                              void* d_out, int out_size, void* d_ws, size_t ws_size,
                              hipStream_t stream) {
    const float* pred_poses     = (const float*)d_in[0];
    const float* gt_poses       = (const float*)d_in[1];
    const float* pred_conf      = (const float*)d_in[2];
    const float* model_vertices = (const float*)d_in[3];
    const int*   class_ids      = (const int*)d_in[4];
    const int*   sym_mask       = (const int*)d_in[5];
    float* out = (float*)d_out;
    float* ws  = (float*)d_ws;

    const int B     = in_sizes[0] / 7;   // 32
    const int nconf = in_sizes[2];       // B * 1024

    float* add_sum  = ws;            // [B]
    float* adds_sum = ws + B;        // [B]
    float* conf_sum = ws + 2 * B;    // [1]

    init_ws_kernel<<<1, 128, 0, stream>>>(ws, 2 * B + 1);
    addloss_kernel<<<B * 16, 256, 0, stream>>>(pred_poses, gt_poses, model_vertices,
                                               class_ids, add_sum, adds_sum);
    conf_kernel<<<(nconf + 255) / 256, 256, 0, stream>>>(pred_conf, conf_sum, nconf);
    finalize_kernel<<<1, 32, 0, stream>>>(pred_poses, class_ids, sym_mask,
                                          add_sum, adds_sum, conf_sum, out, B, nconf);
}